// Model_52879637348864
// MI455X (gfx1250) — compile-verified
//
#include <hip/hip_runtime.h>

// ---------------------------------------------------------------------------
// Problem constants (from the reference): x[M,K] fp32, weight[N,K] fp32,
// out = 0.75 * x @ colsum(weight), shape [M,1] fp32.
// ---------------------------------------------------------------------------
constexpr int M  = 16384;
constexpr int K  = 8192;
constexpr int N  = 8192;
constexpr int K4 = K / 4;
constexpr int M4 = M / 4;
constexpr float COMBINED_SCALE = 0.75f;

typedef __attribute__((ext_vector_type(2))) float v2f;
typedef __attribute__((ext_vector_type(8))) float v8f;

// ---------------------------------------------------------------------------
// Kernel 1: partial column sums of weight. Grid (K4/256, NS); each block owns
// 1024 contiguous columns (256 float4 lanes) and a stripe of rowsPerSplit rows.
// Fully coalesced 16B/lane streaming reads; global_prefetch_b8 to run ahead.
// Deterministic: fixed serial order within a stripe; stripes reduced in k2.
// ---------------------------------------------------------------------------
__global__ void __launch_bounds__(256)
colsum_partial_kernel(const float4* __restrict__ w4,
                      float4* __restrict__ part4,
                      int rowsPerSplit) {
  int col4 = blockIdx.x * blockDim.x + threadIdx.x;          // [0, K4)
  int n0   = blockIdx.y * rowsPerSplit;
  const float4* p = w4 + (size_t)n0 * K4 + col4;
  float4 acc = make_float4(0.f, 0.f, 0.f, 0.f);
  for (int n = 0; n < rowsPerSplit; ++n) {
    __builtin_prefetch(p + (size_t)8 * K4, 0, 1);            // -> global_prefetch_b8
    float4 v = *p;
    acc.x += v.x; acc.y += v.y; acc.z += v.z; acc.w += v.w;
    p += K4;
  }
  part4[(size_t)blockIdx.y * K4 + col4] = acc;
}

// ---------------------------------------------------------------------------
// Kernel 2: reduce the NS partial stripes into colsum[K]. Tiny, deterministic.
// ---------------------------------------------------------------------------
__global__ void __launch_bounds__(256)
colsum_reduce_kernel(const float4* __restrict__ part4,
                     float4* __restrict__ cs4,
                     int ns) {
  int col4 = blockIdx.x * blockDim.x + threadIdx.x;          // [0, K4)
  float4 acc = make_float4(0.f, 0.f, 0.f, 0.f);
  for (int s = 0; s < ns; ++s) {
    float4 v = part4[(size_t)s * K4 + col4];
    acc.x += v.x; acc.y += v.y; acc.z += v.z; acc.w += v.w;
  }
  cs4[col4] = acc;
}

// ---------------------------------------------------------------------------
// Kernel 3: WMMA GEMV partials.  One wave32 owns 16 rows of x.
// Per step of 4 k-values:
//   A (16x4 f32, 2 VGPRs): lane l -> row = l%16, h = l/16;
//     (V0,V1) = x[row][k+2h], x[row][k+2h+1]  == one per-lane b64 load.
//   B (4x16 f32, 2 VGPRs): colsum chunk broadcast across all 16 columns,
//     (V0,V1) = cs[k+2h], cs[k+2h+1] -> uniform s_load_b128 + 2 cndmask.
//   D = A*B + C accumulates the row-dots in every column; we read column 0
//   from lane 0 (rows 0..7 in V0..V7) and lane 16 (rows 8..15).
// EXEC is all-ones through the WMMA loop (full blocks, no divergence).
// K is split KS ways (blockIdx.y) for occupancy; partials combined in k4.
// ---------------------------------------------------------------------------
__global__ void __launch_bounds__(256)
gemv_wmma_kernel(const float* __restrict__ x,
                 const float* __restrict__ colsum,
                 float* __restrict__ ybase,     // KS slabs of M floats (or d_out if KS==1)
                 int kChunk, float outScale) {
  const int lane = threadIdx.x & 31;
  const int wave = threadIdx.x >> 5;
  const int r    = lane & 15;
  const int h    = lane >> 4;
  const int m0   = (blockIdx.x * 8 + wave) * 16;

  const float* xrow = x + (size_t)(m0 + r) * K;
  const int kbeg = blockIdx.y * kChunk;
  const int kend = kbeg + kChunk;
  const int koff = 2 * h;

  v8f c = {0.f, 0.f, 0.f, 0.f, 0.f, 0.f, 0.f, 0.f};

#pragma unroll 4
  for (int k = kbeg; k < kend; k += 4) {
    v2f a = *(const v2f*)(xrow + k + koff);                  // per-lane global_load_b64
    float4 cs = *(const float4*)(colsum + k);                // uniform -> s_load_b128
    v2f b;
    b.x = h ? cs.z : cs.x;
    b.y = h ? cs.w : cs.y;
    // v_wmma_f32_16x16x4_f32: D = A*B + C  (fp32-exact, RNE)
    c = __builtin_amdgcn_wmma_f32_16x16x4_f32(
        /*neg_a=*/false, a, /*neg_b=*/false, b,
        /*c_mod=*/(short)0, c, /*reuse_a=*/false, /*reuse_b=*/false);
  }

  // C/D layout: VGPR v, lanes 0-15 -> M=v, N=lane; lanes 16-31 -> M=v+8.
  float* ydst = ybase + (size_t)blockIdx.y * M + m0;
  if (lane == 0) {
#pragma unroll
    for (int v = 0; v < 8; ++v) ydst[v] = outScale * c[v];
  } else if (lane == 16) {
#pragma unroll
    for (int v = 0; v < 8; ++v) ydst[8 + v] = outScale * c[v];
  }
}

// ---------------------------------------------------------------------------
// Kernel 4: combine the KS k-split partials, apply 0.75 scale. Deterministic.
// ---------------------------------------------------------------------------
__global__ void __launch_bounds__(256)
combine_kernel(const float4* __restrict__ ypart4,
               float4* __restrict__ out4,
               int ks) {
  int i = blockIdx.x * blockDim.x + threadIdx.x;             // [0, M4)
  float4 acc = make_float4(0.f, 0.f, 0.f, 0.f);
  for (int s = 0; s < ks; ++s) {
    float4 v = ypart4[(size_t)s * M4 + i];
    acc.x += v.x; acc.y += v.y; acc.z += v.z; acc.w += v.w;
  }
  float4 o;
  o.x = COMBINED_SCALE * acc.x; o.y = COMBINED_SCALE * acc.y;
  o.z = COMBINED_SCALE * acc.z; o.w = COMBINED_SCALE * acc.w;
  out4[i] = o;
}

// ---------------------------------------------------------------------------
// Launch: deterministic 4-stage pipeline, all on `stream`, scratch in d_ws.
// ws layout (floats): [ NS*K partials | K colsum | KS*M y-partials ]
// Tiered (NS,KS) selection so we never overrun ws_size.
// ---------------------------------------------------------------------------
extern "C" void kernel_launch(void* const* d_in, const int* in_sizes, int n_in,
                              void* d_out, int out_size, void* d_ws, size_t ws_size,
                              hipStream_t stream) {
  const float* x = (const float*)d_in[0];
  const float* w = (const float*)d_in[1];
  float* out = (float*)d_out;
  float* ws  = (float*)d_ws;

  auto need = [](int ns, int ks) -> size_t {
    size_t f = (size_t)(ns + 1) * K + (ks > 1 ? (size_t)ks * M : 0);
    return f * sizeof(float);
  };

  int NS = 64, KS = 8;                         // ~2.6 MB of ws, best occupancy
  if (need(NS, KS) > ws_size) { NS = 16; KS = 8; }
  if (need(NS, KS) > ws_size) { NS = 4;  KS = 4; }
  if (need(NS, KS) > ws_size) { NS = 1;  KS = 1; }

  float* part   = ws;                          // NS * K
  float* colsum = ws + (size_t)NS * K;         // K
  float* ypart  = colsum + K;                  // KS * M (only when KS > 1)

  // Stage 1: partial column sums of weight (reads 256 MB, coalesced).
  colsum_partial_kernel<<<dim3(K4 / 256, NS), 256, 0, stream>>>(
      (const float4*)w, (float4*)part, N / NS);

  // Stage 2: reduce stripes -> colsum[K].
  colsum_reduce_kernel<<<dim3(K4 / 256), 256, 0, stream>>>(
      (const float4*)part, (float4*)colsum, NS);

  // Stage 3 (+4): WMMA GEMV over x (reads 512 MB).
  if (KS > 1) {
    gemv_wmma_kernel<<<dim3(M / 128, KS), 256, 0, stream>>>(
        x, colsum, ypart, K / KS, 1.0f);
    combine_kernel<<<dim3(M4 / 256), 256, 0, stream>>>(
        (const float4*)ypart, (float4*)out, KS);
  } else {
    gemv_wmma_kernel<<<dim3(M / 128, 1), 256, 0, stream>>>(
        x, colsum, out, K, COMBINED_SCALE);
  }
}